// GaussianDecimator_29575144800668
// MI455X (gfx1250) — compile-verified
//
#include <hip/hip_runtime.h>

// FIR anti-alias + decimate-by-2 as a banded implicit GEMM on the CDNA5
// matrix pipe (V_WMMA_F32_16X16X4_F32, wave32).
//
//   y[b, i0+j] = sum_s A[j,s] * B[s,b]
//     A[j,s] = h[s - 2j]            (16 x 656 banded filter matrix, from LDS)
//     B[s,b] = x[b, 2*i0 + s - 312] (per-batch sliding windows, from LDS)
//
// D tile (16x16) = 16 output positions x 16 batches.

typedef __attribute__((ext_vector_type(2))) float v2f;
typedef __attribute__((ext_vector_type(8))) float v8f;

#define T_IN    1048576          // samples per batch
#define T_OUT   524288           // outputs per batch (decimate by 2)
#define NB      16               // batches
#define NTAPS   625
#define HALFK   312
#define KTOT    656              // padded K extent (max s = 624 + 2*15 = 654), /4 = 164 steps
#define QSTEPS  (KTOT / 4)
#define WAVES   8
#define TPW     4                // tiles per wave (amortizes A-fragment loads 4x)
#define TILES   (WAVES * TPW)    // 32 tiles of 16 outputs per block
#define POSB    (TILES * 16)     // 512 output positions per block
#define WIN     (2 * (POSB - 16) + KTOT)   // 1648 x-samples per batch per block
#define XSTR    1650             // LDS row stride: even (8B align), mod 64 = 50 -> bank-spread
#define HPAD    32               // zero pad on both sides of h in LDS
#define HSZ     704              // 32 + 625 + pad -> covers index range [2, 687]

__global__ __launch_bounds__(256) void GaussianDecimator_fir_wmma(
    const float* __restrict__ x,     // (16, 1, 1048576) f32
    const float* __restrict__ h,     // (625,) f32
    float* __restrict__ out)         // (16, 1, 524288) f32
{
    extern __shared__ float smem[];
    float* shx = smem;                 // NB * XSTR floats
    float* shh = smem + NB * XSTR;     // HSZ floats

    const int tid = threadIdx.x;
    const int blk = blockIdx.x;
    const long xg0 = (long)2 * blk * POSB - HALFK;   // x index of window sample s=0

    // ---- stage filter into LDS, zero-padded both sides ----
    for (int i = tid; i < HSZ; i += 256) {
        const int k = i - HPAD;
        shh[i] = (k >= 0 && k < NTAPS) ? h[k] : 0.0f;
    }

    // ---- stage 16 per-batch x windows into LDS ----
    if (xg0 >= 0 && xg0 + WIN <= T_IN) {
        // interior block: fully in-bounds, 16B-aligned (xg0 = 1024*blk - 312, mult of 4)
        const int NV = WIN / 4;  // 412 float4 per batch
        for (int i = tid; i < NB * NV; i += 256) {
            const int b  = i / NV;
            const int s4 = i - b * NV;
            const float4 v = *(const float4*)(x + (long)b * T_IN + xg0 + 4 * s4);
            float* d = shx + b * XSTR + 4 * s4;
            d[0] = v.x; d[1] = v.y; d[2] = v.z; d[3] = v.w;
        }
    } else {
        // boundary block: predicated scalar loads, zero outside [0, T_IN)
        for (int i = tid; i < NB * WIN; i += 256) {
            const int b = i / WIN;
            const int s = i - b * WIN;
            const long gx = xg0 + s;
            shx[b * XSTR + s] = (gx >= 0 && gx < T_IN) ? x[(long)b * T_IN + gx] : 0.0f;
        }
    }
    __syncthreads();

    const int wave = tid >> 5;
    const int lane = tid & 31;
    const int half = lane >> 4;   // K-half select (A/B fragments split K across lane halves)
    const int l    = lane & 15;

    v8f acc[TPW] = {};            // 4 accumulator tiles, 8 VGPRs each

    // A fragment, step q: lane l<16 holds (h[4q-2l], h[4q+1-2l]); lanes 16-31 hold K+2,K+3.
    // Base index HPAD + 2*half - 2*l >= 2, always inside the zero-padded LDS copy.
    const float* hA = shh + HPAD + 2 * half - 2 * l;
    // B fragment, step q, tile t: lane l holds rows s=4q(+2*half)..+1 of batch column l.
    const float* xB = shx + l * XSTR + 2 * half;

    for (int q = 0; q < QSTEPS; ++q) {
        const int k0 = 4 * q;
        const v2f a = *(const v2f*)(hA + k0);          // ds_load_b64 (8B aligned)
        #pragma unroll
        for (int t = 0; t < TPW; ++t) {
            const int tile = wave * TPW + t;
            const v2f b = *(const v2f*)(xB + 32 * tile + k0);  // ds_load_b64
            // D = A(16x4 f32) * B(4x16 f32) + C  -> v_wmma_f32_16x16x4_f32
            acc[t] = __builtin_amdgcn_wmma_f32_16x16x4_f32(
                /*neg_a=*/false, a, /*neg_b=*/false, b,
                /*c_mod=*/(short)0, acc[t],
                /*reuse_a=*/false, /*reuse_b=*/false);
        }
    }

    // ---- store: C/D layout -> VGPR v, lanes 0-15: (j=v, batch=l); lanes 16-31: (j=v+8, batch=l)
    #pragma unroll
    for (int t = 0; t < TPW; ++t) {
        const int i0 = blk * POSB + (wave * TPW + t) * 16;
        const float* av = (const float*)&acc[t];
        #pragma unroll
        for (int v = 0; v < 8; ++v) {
            out[(long)l * T_OUT + i0 + v + 8 * half] = av[v];
        }
    }
}

extern "C" void kernel_launch(void* const* d_in, const int* in_sizes, int n_in,
                              void* d_out, int out_size, void* d_ws, size_t ws_size,
                              hipStream_t stream) {
    const float* x = (const float*)d_in[0];   // (16,1,1048576) f32
    const float* h = (const float*)d_in[1];   // (625,) f32
    float* out = (float*)d_out;               // (16,1,524288) f32

    const size_t shmem = (size_t)(NB * XSTR + HSZ) * sizeof(float);  // ~108 KB of 320 KB WGP LDS
    const dim3 grid(T_OUT / POSB);   // 1024 blocks
    const dim3 block(256);           // 8 wave32 waves

    hipLaunchKernelGGL(GaussianDecimator_fir_wmma, grid, block, shmem, stream, x, h, out);
}